// WellTemparedMetaDynamics_22814866276450
// MI455X (gfx1250) — compile-verified
//
#include <hip/hip_runtime.h>
#include <stdint.h>

// ---------------- problem constants ----------------
#define N_BCH 64
#define N_DIM 64
#define N_COL 32
#define N_HIL 50000

// ---------------- decomposition --------------------
#define NCHUNK 625                   // blocks in kernel 1; 625*80 = 50000 hills
#define HPB    (N_HIL / NCHUNK)      // 80 hills per block
#define TH     4                     // hills per LDS tile
#define NTILE  (HPB / TH)            // 20 tiles per block
#define ROWF   36                    // padded row length in floats (144 B, 16B-aligned rows)
#define TILE_ROWS   (TH * N_BCH)     // 256 rows per tile (one per thread)
#define TILE_FLOATS (TILE_ROWS * ROWF)   // 9216 floats = 36864 B per cen/prc buffer
#define TILE_BYTES  (TILE_FLOATS * 4)    // 36864
#define TILE_ELEMS  (TH * N_BCH * N_COL) // 8192 floats of real payload per tile

// Issue one tile's async loads: 8x b128 for cen, 8x b128 for prc, 1x b32 for hgt
// per thread => 17 async ops per wave per tile (uniform across all waves, so
// s_wait_asynccnt 17 retires exactly one tile while the next streams).
__device__ __forceinline__ void issue_tile(const float* cen_t, const float* prc_t,
                                           const float* hgt_t,
                                           unsigned l_cen, unsigned l_prc, unsigned l_hgt,
                                           int t) {
  const unsigned gb = (unsigned)t * 16u;                                   // global byte offset
  const unsigned lb = (unsigned)(t >> 3) * 144u + (unsigned)(t & 7) * 16u; // padded LDS offset
  const uint64_t sc = (uint64_t)(uintptr_t)cen_t;
  const uint64_t sp = (uint64_t)(uintptr_t)prc_t;
  const uint64_t sh = (uint64_t)(uintptr_t)hgt_t;
#pragma unroll
  for (int i = 0; i < 8; ++i) {
    unsigned go = gb + (unsigned)i * 4096u;          // 256 threads * 16 B per sweep
    unsigned lc = l_cen + lb + (unsigned)i * 4608u;  // 32 rows * 144 B per sweep
    unsigned lp = l_prc + lb + (unsigned)i * 4608u;
    asm volatile("global_load_async_to_lds_b128 %0, %1, %2"
                 :: "v"(lc), "v"(go), "s"(sc) : "memory");
    asm volatile("global_load_async_to_lds_b128 %0, %1, %2"
                 :: "v"(lp), "v"(go), "s"(sp) : "memory");
  }
  unsigned gh = (unsigned)t * 4u;           // 256 threads * 4 B = 1 KB hgt tile
  unsigned lh = l_hgt + (unsigned)t * 4u;
  asm volatile("global_load_async_to_lds_b32 %0, %1, %2"
               :: "v"(lh), "v"(gh), "s"(sh) : "memory");
}

__global__ __launch_bounds__(256) void k_hills(
    const float* __restrict__ col_var, const float* __restrict__ mtd_cen,
    const float* __restrict__ mtd_prc, const float* __restrict__ mtd_hgt,
    const float* __restrict__ pbc, const int* __restrict__ msk,
    float* __restrict__ ws)
{
  __shared__ __align__(16) float s_cen[2][TILE_FLOATS];
  __shared__ __align__(16) float s_prc[2][TILE_FLOATS];
  __shared__ __align__(16) float s_hgt[2][TILE_ROWS];
  __shared__ __align__(16) float s_pcl[N_COL];
  __shared__ __align__(16) float s_ivp[N_COL];
  __shared__ float s_red[256];

  const int t   = threadIdx.x;
  const int blk = blockIdx.x;
  const int b   = t & 63;             // batch owned by this thread
  const int h0  = blk * HPB;          // first hill of this block

  if (t < N_COL) {
    float p = pbc[t];
    bool pos = p > 0.0f;
    s_pcl[t] = pos ? p : 0.0f;
    s_ivp[t] = pos ? 1.0f / p : 0.0f;
  }

  // col row for this batch, kept in registers for the whole hill stream
  float cr[N_COL];
#pragma unroll
  for (int c = 0; c < N_COL; ++c) cr[c] = col_var[b * N_DIM + msk[c]];

  // raw LDS byte addresses for the async engine (LDS aperture has zero low bits)
  const unsigned l_cen0 = (unsigned)(uintptr_t)&s_cen[0][0];
  const unsigned l_prc0 = (unsigned)(uintptr_t)&s_prc[0][0];
  const unsigned l_hgt0 = (unsigned)(uintptr_t)&s_hgt[0][0];

  const float* cen_base = mtd_cen + (size_t)h0 * (N_BCH * N_COL);
  const float* prc_base = mtd_prc + (size_t)h0 * (N_BCH * N_COL);
  const float* hgt_base = mtd_hgt + (size_t)h0 * N_BCH;

  // prologue: fill both buffers
  issue_tile(cen_base, prc_base, hgt_base, l_cen0, l_prc0, l_hgt0, t);
  issue_tile(cen_base + TILE_ELEMS, prc_base + TILE_ELEMS, hgt_base + TH * N_BCH,
             l_cen0 + TILE_BYTES, l_prc0 + TILE_BYTES, l_hgt0 + 1024u, t);

  float eng = 0.0f;
  for (int i = 0; i < NTILE; ++i) {
    // retire tile i (the 17 async ops of tile i+1 may remain in flight)
    if (i + 1 < NTILE) { asm volatile("s_wait_asynccnt 17" ::: "memory"); }
    else               { asm volatile("s_wait_asynccnt 0"  ::: "memory"); }
    __syncthreads();   // all waves' async data for tile i now visible

    const int bf = i & 1;
    const float4* cen_row = (const float4*)&s_cen[bf][t * ROWF];  // row t = (hill, batch)
    const float4* prc_row = (const float4*)&s_prc[bf][t * ROWF];
    float acc = 0.0f;
#pragma unroll
    for (int q = 0; q < 8; ++q) {
      float4 ce = cen_row[q];
      float4 pr = prc_row[q];
      float4 pb = ((const float4*)s_pcl)[q];   // broadcast reads
      float4 ip = ((const float4*)s_ivp)[q];
      float d, r;
      d = cr[q*4+0] - ce.x; r = rintf(d*ip.x); d = fmaf(-pb.x, r, d); acc = fmaf(pr.x*d, d, acc);
      d = cr[q*4+1] - ce.y; r = rintf(d*ip.y); d = fmaf(-pb.y, r, d); acc = fmaf(pr.y*d, d, acc);
      d = cr[q*4+2] - ce.z; r = rintf(d*ip.z); d = fmaf(-pb.z, r, d); acc = fmaf(pr.z*d, d, acc);
      d = cr[q*4+3] - ce.w; r = rintf(d*ip.w); d = fmaf(-pb.w, r, d); acc = fmaf(pr.w*d, d, acc);
    }
    eng += s_hgt[bf][t] * __expf(-0.5f * acc);

    __syncthreads();   // everyone done reading buffer bf before it is refilled
    if (i + 2 < NTILE) {
      const int j = i + 2;
      issue_tile(cen_base + (size_t)j * TILE_ELEMS,
                 prc_base + (size_t)j * TILE_ELEMS,
                 hgt_base + (size_t)j * (TH * N_BCH),
                 l_cen0 + (unsigned)bf * TILE_BYTES,
                 l_prc0 + (unsigned)bf * TILE_BYTES,
                 l_hgt0 + (unsigned)bf * 1024u, t);
    }
  }

  // fold the 4 hill-groups into 64 per-batch partials (deterministic, no atomics)
  s_red[t] = eng;
  __syncthreads();
  if (t < N_BCH) {
    float v = s_red[t] + s_red[64 + t] + s_red[128 + t] + s_red[192 + t];
    ws[(size_t)blk * N_BCH + t] = v;
  }
}

__global__ __launch_bounds__(256) void k_final(
    const float* __restrict__ ws, const float* __restrict__ col_var,
    const int* __restrict__ msk, const float* __restrict__ prc,
    const float* __restrict__ kbt, const float* __restrict__ hgt,
    const float* __restrict__ gam, float* __restrict__ out)
{
  const int t = threadIdx.x;
  for (int idx = t; idx < N_BCH * N_COL; idx += 256) {
    int bb = idx >> 5, j = idx & 31;
    out[idx] = col_var[bb * N_DIM + msk[j]];          // col
    out[N_BCH * N_COL + idx] = prc[j];                // prc_new
  }
  if (t < N_BCH) {
    float eng = 0.0f;
    for (int k = 0; k < NCHUNK; ++k) eng += ws[(size_t)k * N_BCH + t];
    float kb  = kbt[t];
    float det = gam[0] * kb - kb;
    out[2 * N_BCH * N_COL + t] = hgt[0] * __expf(-eng / det);   // hgt_wt
  }
}

extern "C" void kernel_launch(void* const* d_in, const int* in_sizes, int n_in,
                              void* d_out, int out_size, void* d_ws, size_t ws_size,
                              hipStream_t stream) {
  const float* col_var = (const float*)d_in[0];
  const float* mtd_cen = (const float*)d_in[1];
  const float* mtd_prc = (const float*)d_in[2];
  const float* mtd_hgt = (const float*)d_in[3];
  const float* kbt     = (const float*)d_in[4];
  const float* prc     = (const float*)d_in[5];
  const float* hgt     = (const float*)d_in[6];
  const float* gam     = (const float*)d_in[7];
  const float* pbc     = (const float*)d_in[8];
  const int*   msk     = (const int*)d_in[9];
  float* ws  = (float*)d_ws;     // 625 * 64 * 4 = 160 KB of partials
  float* out = (float*)d_out;    // 4160 floats

  k_hills<<<NCHUNK, 256, 0, stream>>>(col_var, mtd_cen, mtd_prc, mtd_hgt, pbc, msk, ws);
  k_final<<<1, 256, 0, stream>>>(ws, col_var, msk, prc, kbt, hgt, gam, out);
}